// GraphNeuralODE_7035156431296
// MI455X (gfx1250) — compile-verified
//
#include <hip/hip_runtime.h>
#include <hip/hip_bf16.h>
#include <math.h>

typedef __attribute__((ext_vector_type(2))) float v2f;
typedef __attribute__((ext_vector_type(8))) float v8f;

#define N_NODES   4096
#define T_IN      24
#define HORIZON   10
#define HID       64
#define NTHREADS  256
#define PER_THR   (N_NODES / NTHREADS)   // 16

__launch_bounds__(NTHREADS, 1)
__global__ void gnode_kernel(const float* __restrict__ x,
                             const float* __restrict__ W1,
                             const float* __restrict__ b1,
                             const float* __restrict__ W2,
                             const float* __restrict__ b2,
                             const float* __restrict__ W3,
                             const float* __restrict__ b3,
                             float* __restrict__ out)
{
    __shared__ float sW2[HID * HID];   // row-major [i][j]
    __shared__ float sw1[HID];
    __shared__ float sb1[HID];
    __shared__ float sb2[HID];
    __shared__ float sW3[HID];
    __shared__ float sh1[HID];
    __shared__ float sred[NTHREADS / 32];
    __shared__ float sdelta[HORIZON];

    const int tid  = threadIdx.x;
    const int lane = tid & 31;
    const int wave = tid >> 5;

    // Stage weights into LDS (16 KB + small vectors; WGP has 320 KB).
    for (int i = tid; i < HID * HID; i += NTHREADS) sW2[i] = W2[i];
    if (tid < HID) {
        sw1[tid] = W1[tid];
        sb1[tid] = b1[tid];
        sb2[tid] = b2[tid];
        sW3[tid] = W3[tid];
    }
    if (tid == 0) sdelta[0] = 0.0f;

    // y0 = x[:, :, -1]; keep 16 values per thread in registers, reduce the sum.
    float y0v[PER_THR];
    float psum = 0.0f;
#pragma unroll
    for (int j = 0; j < PER_THR; ++j) {
        const int n = tid + j * NTHREADS;
        const float v = x[n * T_IN + (T_IN - 1)];
        y0v[j] = v;
        psum += v;
    }
#pragma unroll
    for (int off = 16; off > 0; off >>= 1) psum += __shfl_xor(psum, off, 32);
    if (lane == 0) sred[wave] = psum;
    __syncthreads();

    if (wave == 0) {
        // Finish the mean reduction (broadcast to all 32 lanes).
        float tot = (lane < NTHREADS / 32) ? sred[lane] : 0.0f;
#pragma unroll
        for (int off = 16; off > 0; off >>= 1) tot += __shfl_xor(tot, off, 32);
        float s = tot * (1.0f / (float)N_NODES);   // scalar ODE state = mean(y)

        const float bb3 = b3[0];
        const float DTc = 10.0f / 9.0f;            // H/(H-1)

        const int   half  = lane >> 4;             // 0: lanes 0-15, 1: lanes 16-31
        const int   col   = lane & 15;
        const float amask = (col == 0) ? 1.0f : 0.0f;  // keep only A row M=0 live

        // g(s) = W3^T tanh(W2^T tanh(s*w1+b1) + b2) + b3, matvec via WMMA f32 16x16x4.
        auto gfun = [&](float sv) -> float {
            // h1 = tanh(sv*w1 + b1): 64 values, 2 per lane, staged through LDS.
            sh1[lane]      = tanhf(fmaf(sv, sw1[lane],      sb1[lane]));
            sh1[lane + 32] = tanhf(fmaf(sv, sw1[lane + 32], sb1[lane + 32]));
            __builtin_amdgcn_wave_barrier();   // keep same-wave LDS write->read order

            v8f acc0 = {}, acc1 = {}, acc2 = {}, acc3 = {};
#pragma unroll
            for (int kb = 0; kb < 16; ++kb) {
                const int r0 = 4 * kb + 2 * half;
                // A (16x4 f32): A[0][k] = h1[4kb+k]; rows 1-15 zeroed via amask.
                // Layout: lane = M + 16*(K>=2), vgpr = K&1  ->  lanes 0/16 carry row 0.
                v2f a;
                a.x = sh1[r0]     * amask;
                a.y = sh1[r0 + 1] * amask;
                // B (4x16 f32): B[k][n] = W2[4kb+k][16t+n]; vgpr = K&1, lane = n + 16*(K>=2).
                const float* wp = &sW2[r0 * HID + col];
                v2f b0, b1f, b2f, b3f;
                b0.x  = wp[0];   b0.y  = wp[HID];
                b1f.x = wp[16];  b1f.y = wp[HID + 16];
                b2f.x = wp[32];  b2f.y = wp[HID + 32];
                b3f.x = wp[48];  b3f.y = wp[HID + 48];
                // 4 independent accumulator chains interleave to hide XDL latency.
                acc0 = __builtin_amdgcn_wmma_f32_16x16x4_f32(false, a, false, b0,  (short)0, acc0, false, false);
                acc1 = __builtin_amdgcn_wmma_f32_16x16x4_f32(false, a, false, b1f, (short)0, acc1, false, false);
                acc2 = __builtin_amdgcn_wmma_f32_16x16x4_f32(false, a, false, b2f, (short)0, acc2, false, false);
                acc3 = __builtin_amdgcn_wmma_f32_16x16x4_f32(false, a, false, b3f, (short)0, acc3, false, false);
            }
            __builtin_amdgcn_wave_barrier();

            // D row 0 lives in element 0 on lanes 0-15 (N = lane). Apply b2, tanh, dot W3.
            float partial = 0.0f;
            if (lane < 16) {
                partial  = tanhf(acc0[0] + sb2[ 0 + col]) * sW3[ 0 + col];
                partial += tanhf(acc1[0] + sb2[16 + col]) * sW3[16 + col];
                partial += tanhf(acc2[0] + sb2[32 + col]) * sW3[32 + col];
                partial += tanhf(acc3[0] + sb2[48 + col]) * sW3[48 + col];
            }
#pragma unroll
            for (int off = 16; off > 0; off >>= 1) partial += __shfl_xor(partial, off, 32);
            return partial + bb3;
        };

        // torchdiffeq 'rk4' (3/8 rule), 9 fixed steps; keep the stage loop rolled
        // so the binary carries one 64-WMMA matvec body.
        float cum = 0.0f;
#pragma unroll 1
        for (int step = 0; step < HORIZON - 1; ++step) {
            float k0 = 0.0f, k1v = 0.0f, k2v = 0.0f;
#pragma unroll 1
            for (int st = 0; st < 4; ++st) {
                float arg;
                if      (st == 0) arg = s;
                else if (st == 1) arg = fmaf(DTc * (1.0f / 3.0f), k0, s);
                else if (st == 2) arg = s + DTc * (k1v - (1.0f / 3.0f) * k0);
                else              arg = s + DTc * (k0 - k1v + k2v);
                const float kv = gfun(arg);
                if      (st == 0) k0  = kv;
                else if (st == 1) k1v = kv;
                else if (st == 2) k2v = kv;
                else {
                    const float inc = DTc * (k0 + 3.0f * k1v + 3.0f * k2v + kv) * 0.125f;
                    s   += inc;
                    cum += inc;
                    if (lane == 0) sdelta[step + 1] = cum;
                }
            }
        }
    }
    __syncthreads();

    // out[0, n, t] = y0[n] + delta[t]   (flat: n*HORIZON + t)
    float dl[HORIZON];
#pragma unroll
    for (int t = 0; t < HORIZON; ++t) dl[t] = sdelta[t];
#pragma unroll
    for (int j = 0; j < PER_THR; ++j) {
        const int n = tid + j * NTHREADS;
        const float base = y0v[j];
#pragma unroll
        for (int t = 0; t < HORIZON; ++t) out[n * HORIZON + t] = base + dl[t];
    }
}

extern "C" void kernel_launch(void* const* d_in, const int* in_sizes, int n_in,
                              void* d_out, int out_size, void* d_ws, size_t ws_size,
                              hipStream_t stream) {
    const float* x  = (const float*)d_in[0];
    const float* W1 = (const float*)d_in[1];
    const float* b1 = (const float*)d_in[2];
    const float* W2 = (const float*)d_in[3];
    const float* b2 = (const float*)d_in[4];
    const float* W3 = (const float*)d_in[5];
    const float* b3 = (const float*)d_in[6];
    float* out = (float*)d_out;
    gnode_kernel<<<1, NTHREADS, 0, stream>>>(x, W1, b1, W2, b2, W3, b3, out);
}